// RotationPrior_88175678587354
// MI455X (gfx1250) — compile-verified
//
#include <hip/hip_runtime.h>

// CDNA5 / gfx1250: wave32, WMMA 16x16x4 f32.
typedef __attribute__((ext_vector_type(2))) float v2f;
typedef __attribute__((ext_vector_type(8))) float v8f;

#if defined(__has_builtin)
#if __has_builtin(__builtin_amdgcn_wmma_f32_16x16x4_f32)
#define RP_HAVE_WMMA_F32X4 1
#endif
#endif

#define WAVES_PER_BLOCK 8

// ---------------------------------------------------------------------------
// Kernel 1: out = pos (reference starts from pos and scatters updates into it)
// ---------------------------------------------------------------------------
__global__ void rp_copy_kernel(const float* __restrict__ src,
                               float* __restrict__ dst, long long n) {
  long long n4 = n >> 2;
  long long tid = (long long)blockIdx.x * blockDim.x + threadIdx.x;
  long long stride = (long long)gridDim.x * blockDim.x;
  const float4* s4 = (const float4*)src;
  float4* d4 = (float4*)dst;
  for (long long j = tid; j < n4; j += stride) d4[j] = s4[j];
  for (long long j = (n4 << 2) + tid; j < n; j += stride) dst[j] = src[j];
}

__device__ __forceinline__ long long rp_lower_bound(
    const long long* __restrict__ a, long long n, long long key) {
  long long lo = 0, hi = n;
  while (lo < hi) {
    long long mid = (lo + hi) >> 1;
    if (a[mid] < key) lo = mid + 1; else hi = mid;
  }
  return lo;
}

// ---------------------------------------------------------------------------
// Kernel 2: one wave (32 lanes) per domain.
//   pass 1: wave-strided centroid reduction over the sorted run [lo,hi)
//   pass 2: Rodrigues rotation of 16-node tiles via v_wmma_f32_16x16x4_f32:
//           A(16x4) = R (rows 0..2, rest 0)   -- wave-uniform per domain
//           B(4x16) = 16 relative vectors (column n = node t+n)
//           D(16x16) rows 0..2 = rotated vectors; lane n reads D[0..2][n]
// ---------------------------------------------------------------------------
__global__ __launch_bounds__(WAVES_PER_BLOCK * 32)
void rp_rotate_kernel(const float* __restrict__ pos,
                      const float* __restrict__ axes,
                      const float* __restrict__ angles,
                      const long long* __restrict__ dom,
                      const long long* __restrict__ nidx,
                      float* __restrict__ out,
                      long long n_edge, int n_domain) {
  int wave = blockIdx.x * WAVES_PER_BLOCK + (threadIdx.x >> 5);
  if (wave >= n_domain) return;                 // wave-uniform exit
  int lane = threadIdx.x & 31;
  long long d = wave;

  long long lo = rp_lower_bound(dom, n_edge, d);
  long long hi = rp_lower_bound(dom, n_edge, d + 1);
  if (lo >= hi) return;                         // wave-uniform exit

  // ---- pass 1: centroid (segment mean) ----
  float sx = 0.f, sy = 0.f, sz = 0.f, sc = 0.f;
  for (long long i = lo + lane; i < hi; i += 32) {
    long long node = nidx[i];
    const float* p = pos + 3 * node;
    sx += p[0]; sy += p[1]; sz += p[2]; sc += 1.0f;
  }
  for (int m = 16; m >= 1; m >>= 1) {           // wave32 butterfly reduce
    sx += __shfl_xor(sx, m, 32);
    sy += __shfl_xor(sy, m, 32);
    sz += __shfl_xor(sz, m, 32);
    sc += __shfl_xor(sc, m, 32);
  }
  float inv = 1.0f / fmaxf(sc, 1.0f);
  float cx = sx * inv, cy = sy * inv, cz = sz * inv;

  // ---- rotation matrix (wave-uniform) ----
  float ax = axes[3 * d], ay = axes[3 * d + 1], az = axes[3 * d + 2];
  float rn = 1.0f / sqrtf(ax * ax + ay * ay + az * az);
  float kx = ax * rn, ky = ay * rn, kz = az * rn;
  float th = angles[d];
  float ct = cosf(th), st = sinf(th), oc = 1.0f - ct;
  float R00 = ct + oc * kx * kx;
  float R01 = oc * kx * ky - st * kz;
  float R02 = oc * kx * kz + st * ky;
  float R10 = oc * ky * kx + st * kz;
  float R11 = ct + oc * ky * ky;
  float R12 = oc * ky * kz - st * kx;
  float R20 = oc * kz * kx - st * ky;
  float R21 = oc * kz * ky + st * kx;
  float R22 = ct + oc * kz * kz;

  int m = lane & 15;
  bool lohalf = lane < 16;

#if RP_HAVE_WMMA_F32X4
  // A 16x4 f32 VGPR layout: lane m<16 -> {A[m][0], A[m][1]},
  //                         lane m+16 -> {A[m][2], A[m][3]=0}
  float col0 = (m == 0) ? R00 : (m == 1) ? R10 : (m == 2) ? R20 : 0.0f;
  float col1 = (m == 0) ? R01 : (m == 1) ? R11 : (m == 2) ? R21 : 0.0f;
  float col2 = (m == 0) ? R02 : (m == 1) ? R12 : (m == 2) ? R22 : 0.0f;
  v2f A;
  A.x = lohalf ? col0 : col2;
  A.y = lohalf ? col1 : 0.0f;

  // ---- pass 2: 16-node tiles through the matrix engine ----
  for (long long t = lo; t < hi; t += 16) {
    long long i = t + m;
    bool valid = i < hi;
    // B 4x16 f32 layout: lane n<16 -> {B[0][n]=x_n, B[1][n]=y_n},
    //                    lane 16+n -> {B[2][n]=z_n, B[3][n]=0}
    float b0 = 0.0f, b1 = 0.0f;
    long long node = 0;
    if (valid) {
      node = nidx[i];
      const float* p = pos + 3 * node;
      if (lohalf) { b0 = p[0] - cx; b1 = p[1] - cy; }
      else        { b0 = p[2] - cz; }
    }
    v2f B; B.x = b0; B.y = b1;
    v8f C = {};
    // D = A x B (+0): D[m][n] = (R * rel_n)[m]; EXEC is all-1s here.
    v8f D = __builtin_amdgcn_wmma_f32_16x16x4_f32(
        /*neg_a=*/false, A, /*neg_b=*/false, B,
        /*c_mod=*/(short)0, C, /*reuse_a=*/false, /*reuse_b=*/false);
    if (lohalf && valid) {
      float* o = out + 3 * node;
      o[0] = D[0] + cx;       // D VGPR0, lanes 0-15 = row M=0 (x')
      o[1] = D[1] + cy;       // row M=1 (y')
      o[2] = D[2] + cz;       // row M=2 (z')
    }
  }
#else
  // VALU fallback (should not trigger on gfx1250 toolchains)
  for (long long i = lo + lane; i < hi; i += 32) {
    long long node = nidx[i];
    const float* p = pos + 3 * node;
    float vx = p[0] - cx, vy = p[1] - cy, vz = p[2] - cz;
    float* o = out + 3 * node;
    o[0] = R00 * vx + R01 * vy + R02 * vz + cx;
    o[1] = R10 * vx + R11 * vy + R12 * vz + cy;
    o[2] = R20 * vx + R21 * vy + R22 * vz + cz;
  }
#endif
}

// ---------------------------------------------------------------------------
// Host-side launcher. Sizes come from in_sizes (no device reads: graph-safe).
// Inputs: 0=pos(f32,3N) 1=axes(f32,3D) 2=angles(f32,D)
//         3=domain_index(i64,N) 4=node_index(i64,N) 5=n_domain(scalar)
// ---------------------------------------------------------------------------
extern "C" void kernel_launch(void* const* d_in, const int* in_sizes, int n_in,
                              void* d_out, int out_size, void* d_ws, size_t ws_size,
                              hipStream_t stream) {
  const float* pos        = (const float*)d_in[0];
  const float* axes       = (const float*)d_in[1];
  const float* angles     = (const float*)d_in[2];
  const long long* dom    = (const long long*)d_in[3];
  const long long* nidx   = (const long long*)d_in[4];
  float* out              = (float*)d_out;

  int n_domain    = in_sizes[2];            // angles length == n_domain
  long long n_edge = (long long)in_sizes[3];
  long long n_pos  = (long long)in_sizes[0];

  // Kernel 1: out = pos
  long long n4 = n_pos >> 2;
  long long cblocks = (n4 + 255) / 256;
  if (cblocks < 1) cblocks = 1;
  if (cblocks > 65535) cblocks = 65535;
  rp_copy_kernel<<<(int)cblocks, 256, 0, stream>>>(pos, out, n_pos);

  // Kernel 2: one wave per domain
  int rblocks = (n_domain + WAVES_PER_BLOCK - 1) / WAVES_PER_BLOCK;
  if (rblocks < 1) rblocks = 1;
  rp_rotate_kernel<<<rblocks, WAVES_PER_BLOCK * 32, 0, stream>>>(
      pos, axes, angles, dom, nidx, out, n_edge, n_domain);
}